// CVAE_15161234554994
// MI455X (gfx1250) — compile-verified
//
#include <hip/hip_runtime.h>

// CVAE forward for MI455X (gfx1250). All GEMMs use V_WMMA_F32_16X16X4_F32
// (fp32 in/out, matching the reference). One wave (32 threads) per 16x16
// output tile; EXEC is all-ones as WMMA requires.

typedef float v2f __attribute__((ext_vector_type(2)));
typedef float v8f __attribute__((ext_vector_type(8)));

#define B_  256
#define N_  100000
#define E_  64
#define K_  10
#define H_  512
#define L_  16

__device__ __forceinline__ v8f wmma_f32(v2f a, v2f b, v8f c) {
  // (neg_a, A, neg_b, B, c_mod, C, reuse_a, reuse_b)
  return __builtin_amdgcn_wmma_f32_16x16x4_f32(false, a, false, b, (short)0, c,
                                               false, false);
}

// ---------------------------------------------------------------- zero scratch
__global__ void k_zero(float* __restrict__ p, int n) {
  int i = blockIdx.x * blockDim.x + threadIdx.x;
  if (i < n) p[i] = 0.f;
}

// ---------------------------------------------- S[b,e] = sum_n mask[b,n]*Ei[n,e]
// cnt[b] = sum_n mask[b,n].  WMMA over (b-tile=16) x (n chunk) x (e=64).
__global__ void k_masked_sum(const int* __restrict__ mask,
                             const float* __restrict__ Ei,
                             float* __restrict__ S, float* __restrict__ cnt) {
  const int lane = threadIdx.x;
  const int row  = lane & 15;
  const int hi   = lane >> 4;          // 0: K={0,1}, 1: K={2,3}
  const int b0   = blockIdx.y * 16;
  const int nbase = blockIdx.x * 500;  // 200 * 500 = 100000

  v8f acc[4] = {{}, {}, {}, {}};
  float csum = 0.f;

  for (int n = nbase; n < nbase + 500; n += 4) {
    const int nA = n + 2 * hi;
    const int* mrow = mask + (b0 + row) * N_ + nA;
    v2f a;
    a.x = (float)mrow[0];
    a.y = (float)mrow[1];
    csum += a.x + a.y;
#pragma unroll
    for (int t = 0; t < 4; ++t) {
      v2f b;
      b.x = Ei[nA * E_ + t * 16 + row];
      b.y = Ei[(nA + 1) * E_ + t * 16 + row];
      acc[t] = wmma_f32(a, b, acc[t]);
    }
  }
#pragma unroll
  for (int t = 0; t < 4; ++t)
#pragma unroll
    for (int r = 0; r < 8; ++r)
      unsafeAtomicAdd(&S[(b0 + r + 8 * hi) * E_ + t * 16 + row], acc[t][r]);
  unsafeAtomicAdd(&cnt[b0 + row], csum);
}

// --------------------------------- c = [S/cnt, emb_resp[label]], FPrior MLP
__global__ void k_prep_fprior(const float* __restrict__ S,
                              const float* __restrict__ cnt,
                              const float* __restrict__ emb_resp,
                              const int* __restrict__ label,
                              const float* __restrict__ pw1, const float* __restrict__ pb1,
                              const float* __restrict__ pw2, const float* __restrict__ pb2,
                              const float* __restrict__ pwm, const float* __restrict__ pbm,
                              const float* __restrict__ pwv, const float* __restrict__ pbv,
                              float* __restrict__ c,
                              float* __restrict__ prior_mean,
                              float* __restrict__ prior_logvar) {
  const int b = blockIdx.x;
  const int t = threadIdx.x;  // 64 threads
  __shared__ float cs[128];
  __shared__ float h1[16];
  __shared__ float h2[32];

  const float inv = 1.0f / cnt[b];
  cs[t]      = S[b * 64 + t] * inv;
  cs[64 + t] = emb_resp[label[b] * 64 + t];
  __syncthreads();
  c[b * 128 + t]      = cs[t];
  c[b * 128 + 64 + t] = cs[64 + t];

  if (t < 16) {
    float s = pb1[t];
    for (int i = 0; i < 128; ++i) s += cs[i] * pw1[i * 16 + t];
    h1[t] = fmaxf(s, 0.f);
  }
  __syncthreads();
  if (t < 32) {
    float s = pb2[t];
    for (int i = 0; i < 16; ++i) s += h1[i] * pw2[i * 32 + t];
    h2[t] = fmaxf(s, 0.f);
  }
  __syncthreads();
  if (t < 16) {
    float m = pbm[t], v = pbv[t];
    for (int i = 0; i < 32; ++i) {
      m += h2[i] * pwm[i * 16 + t];
      v += h2[i] * pwv[i * 16 + t];
    }
    prior_mean[b * 16 + t]   = m;
    prior_logvar[b * 16 + t] = v;
  }
}

// ------------------------------------------ x = [Ei[slate].flatten, c]  [B,768]
__global__ void k_build_x(const float* __restrict__ Ei,
                          const int* __restrict__ slate,
                          const float* __restrict__ c, float* __restrict__ x) {
  const int b = blockIdx.x;
  for (int idx = threadIdx.x; idx < 768; idx += blockDim.x) {
    float v;
    if (idx < 640) {
      const int k = idx >> 6, e = idx & 63;
      v = Ei[slate[b * K_ + k] * E_ + e];
    } else {
      v = c[b * 128 + (idx - 640)];
    }
    x[b * 768 + idx] = v;
  }
}

// ------------------------- generic C = act(A[M,Kd] @ W[Kd,Nd] + bias), WMMA f32
// grid = (Nd/16, M/16), block = 32 (one wave).
__global__ void k_gemm(const float* __restrict__ A, const float* __restrict__ W,
                       const float* __restrict__ bias, float* __restrict__ C,
                       int Kd, int Nd, int relu) {
  const int lane = threadIdx.x;
  const int row  = lane & 15;
  const int hi   = lane >> 4;
  const int n0   = blockIdx.x * 16;
  const int m0   = blockIdx.y * 16;

  v8f acc = {};
  for (int k = 0; k < Kd; k += 4) {
    v2f a = *(const v2f*)(A + (m0 + row) * Kd + k + 2 * hi);
    v2f b;
    b.x = W[(k + 2 * hi) * Nd + n0 + row];
    b.y = W[(k + 2 * hi + 1) * Nd + n0 + row];
    acc = wmma_f32(a, b, acc);
  }
  const float bv = bias[n0 + row];
#pragma unroll
  for (int r = 0; r < 8; ++r) {
    float v = acc[r] + bv;
    if (relu) v = fmaxf(v, 0.f);
    C[(m0 + r + 8 * hi) * Nd + n0 + row] = v;
  }
}

// ---------------------------------------- z = zm + eps*exp(0.5*zlv); dz=[z, c]
__global__ void k_reparam(const float* __restrict__ zm,
                          const float* __restrict__ zlv,
                          const float* __restrict__ eps,
                          const float* __restrict__ c, float* __restrict__ dz) {
  const int b = blockIdx.x;
  const int t = threadIdx.x;  // 160 threads, 144 active
  if (t < 16) {
    dz[b * 144 + t] = zm[b * 16 + t] + eps[b * 16 + t] * expf(0.5f * zlv[b * 16 + t]);
  } else if (t < 144) {
    dz[b * 144 + t] = c[b * 128 + (t - 16)];
  }
}

// ----------------- dot[bk, n] = rec[bk, :64] @ Ei[n, :64]^T   ([2560, 100000])
// A-tile preloaded in 32 VGPRs, 4 N-tiles per wave. Stores are the only HBM
// traffic (~1.02 GB); Ei (25.6 MB) streams from L2.
__global__ void k_dot(const float* __restrict__ rec, const float* __restrict__ Ei,
                      float* __restrict__ out) {
  const int lane = threadIdx.x;
  const int row  = lane & 15;
  const int hi   = lane >> 4;
  const int m0   = blockIdx.y * 16;

  v2f a[16];
#pragma unroll
  for (int kk = 0; kk < 16; ++kk)
    a[kk] = *(const v2f*)(rec + (m0 + row) * E_ + 4 * kk + 2 * hi);

#pragma unroll
  for (int t = 0; t < 4; ++t) {
    const int ntile = blockIdx.x * 4 + t;
    if (ntile * 16 >= N_) return;  // uniform across the wave
    const int ncol = ntile * 16 + row;
    v8f acc = {};
#pragma unroll
    for (int kk = 0; kk < 16; ++kk) {
      v2f b = *(const v2f*)(Ei + ncol * E_ + 4 * kk + 2 * hi);
      acc = wmma_f32(a[kk], b, acc);
    }
#pragma unroll
    for (int r = 0; r < 8; ++r)
      out[(long)(m0 + r + 8 * hi) * N_ + ncol] = acc[r];
  }
}

extern "C" void kernel_launch(void* const* d_in, const int* in_sizes, int n_in,
                              void* d_out, int out_size, void* d_ws, size_t ws_size,
                              hipStream_t stream) {
  // params in setup_inputs() dict order
  const float* Ei       = (const float*)d_in[0];   // [N, 64]
  const float* emb_resp = (const float*)d_in[1];   // [4, 64]
  const float* pw1 = (const float*)d_in[2];  const float* pb1 = (const float*)d_in[3];
  const float* pw2 = (const float*)d_in[4];  const float* pb2 = (const float*)d_in[5];
  const float* pwm = (const float*)d_in[6];  const float* pbm = (const float*)d_in[7];
  const float* pwv = (const float*)d_in[8];  const float* pbv = (const float*)d_in[9];
  const float* ew1 = (const float*)d_in[10]; const float* eb1 = (const float*)d_in[11];
  const float* ew2 = (const float*)d_in[12]; const float* eb2 = (const float*)d_in[13];
  const float* ewm = (const float*)d_in[14]; const float* ebm = (const float*)d_in[15];
  const float* ewv = (const float*)d_in[16]; const float* ebv = (const float*)d_in[17];
  const float* dw1 = (const float*)d_in[18]; const float* db1 = (const float*)d_in[19];
  const float* dw2 = (const float*)d_in[20]; const float* db2 = (const float*)d_in[21];
  const float* dw3 = (const float*)d_in[22]; const float* db3 = (const float*)d_in[23];
  const int*   mask  = (const int*)d_in[24];   // user_repr [B, N]
  const int*   slate = (const int*)d_in[25];   // [B, 10]
  const int*   label = (const int*)d_in[26];   // [B]
  const float* eps   = (const float*)d_in[27]; // [B, 16]

  float* out          = (float*)d_out;
  float* prior_mean   = out;           // [256*16]
  float* prior_logvar = out + 4096;
  float* z_mean       = out + 8192;
  float* z_logvar     = out + 12288;
  float* dot          = out + 16384;   // [2560, 100000]

  float* ws  = (float*)d_ws;
  float* S   = ws;            // 16384
  float* cnt = ws + 16384;    // 256
  float* c   = ws + 16640;    // 32768
  float* x   = ws + 49408;    // 196608
  float* h1  = ws + 246016;   // 131072
  float* h2  = ws + 377088;   // 131072
  float* dz  = ws + 508160;   // 36864
  float* g1  = ws + 545024;   // 131072
  float* g2  = ws + 676096;   // 131072
  float* rec = ws + 807168;   // 163840  (total ~3.9 MB)

  k_zero<<<(16640 + 255) / 256, 256, 0, stream>>>(S, 16640);  // S + cnt
  k_masked_sum<<<dim3(200, 16), 32, 0, stream>>>(mask, Ei, S, cnt);
  k_prep_fprior<<<B_, 64, 0, stream>>>(S, cnt, emb_resp, label,
                                       pw1, pb1, pw2, pb2, pwm, pbm, pwv, pbv,
                                       c, prior_mean, prior_logvar);
  k_build_x<<<B_, 256, 0, stream>>>(Ei, slate, c, x);
  // Encoder
  k_gemm<<<dim3(32, 16), 32, 0, stream>>>(x,  ew1, eb1, h1, 768, 512, 1);
  k_gemm<<<dim3(32, 16), 32, 0, stream>>>(h1, ew2, eb2, h2, 512, 512, 1);
  k_gemm<<<dim3(1, 16),  32, 0, stream>>>(h2, ewm, ebm, z_mean,   512, 16, 0);
  k_gemm<<<dim3(1, 16),  32, 0, stream>>>(h2, ewv, ebv, z_logvar, 512, 16, 0);
  k_reparam<<<B_, 160, 0, stream>>>(z_mean, z_logvar, eps, c, dz);
  // Decoder
  k_gemm<<<dim3(32, 16), 32, 0, stream>>>(dz, dw1, db1, g1, 144, 512, 1);
  k_gemm<<<dim3(32, 16), 32, 0, stream>>>(g1, dw2, db2, g2, 512, 512, 1);
  k_gemm<<<dim3(40, 16), 32, 0, stream>>>(g2, dw3, db3, rec, 512, 640, 1);
  // Scoring einsum: [2560, 64] @ [64, 100000]
  k_dot<<<dim3(1563, 160), 32, 0, stream>>>(rec, Ei, dot);
}